// GIN_20469814133291
// MI455X (gfx1250) — compile-verified
//
#include <hip/hip_runtime.h>
#include <hip/hip_bf16.h>

typedef __attribute__((ext_vector_type(2))) float v2f;
typedef __attribute__((ext_vector_type(8))) float v8f;

static constexpr int kNodes = 50000;
static constexpr int kEdges = 800000;

// ---------------- small helpers ----------------

__global__ void k_zero(float* __restrict__ p, int n) {
  int i = blockIdx.x * blockDim.x + threadIdx.x;
  if (i < n) p[i] = 0.0f;
}

__global__ void k_copy4(const float4* __restrict__ s, float4* __restrict__ d, int n4) {
  int i = blockIdx.x * blockDim.x + threadIdx.x;
  if (i < n4) d[i] = s[i];
}

// ---------------- edge scatter: agg[dst] += x[src], one wave per edge ----------------

__global__ void k_scatter_add(const float* __restrict__ x, float* __restrict__ agg,
                              const int* __restrict__ srcIdx, const int* __restrict__ dstIdx,
                              int nEdges, int C) {
  int gid = blockIdx.x * blockDim.x + threadIdx.x;
  int e = gid >> 5;
  if (e >= nEdges) return;
  int lane = gid & 31;
  int s = srcIdx[e];
  int d = dstIdx[e];
  const float* xs = x + (size_t)s * C;
  float* ad = agg + (size_t)d * C;
  for (int c = lane; c < C; c += 32) atomicAdd(&ad[c], xs[c]);
}

// ---------------- fp32 WMMA GEMM: D = op(A) @ W + bias ----------------
// PREOP=1: A element preprocessed as relu(a*scale[k]+shift[k])  (fused BN+ReLU)
// POSTRELU=1: relu on the output.
// One wave computes a 16x16 tile; wave w in the block owns columns [16w,16w+16).
// Requires M % 16 == 0, K % 4 == 0, Cout == 16 * (blockDim.x/32).

template <int PREOP, int POSTRELU>
__global__ void k_gemm_wmma(const float* __restrict__ A, const float* __restrict__ W,
                            const float* __restrict__ bias,
                            const float* __restrict__ scale, const float* __restrict__ shift,
                            float* __restrict__ D, int K, int Cout) {
  const int wave = threadIdx.x >> 5;
  const int lane = threadIdx.x & 31;
  const int half = lane >> 4;          // 0: lanes 0-15 (k,k+1), 1: lanes 16-31 (k+2,k+3)
  const int l15  = lane & 15;
  const int tileM = blockIdx.x << 4;
  const int tileN = wave << 4;
  const size_t rowA = (size_t)(tileM + l15) * (size_t)K;
  const int colB = tileN + l15;

  v8f c = {};
#pragma unroll 8
  for (int k = 0; k < K; k += 4) {
    const int ka = k + half * 2;
    float2 av = *(const float2*)(A + rowA + ka);   // 8B aligned: ka is even
    v2f a, b;
    if (PREOP) {
      a.x = fmaxf(fmaf(av.x, scale[ka],     shift[ka]),     0.0f);
      a.y = fmaxf(fmaf(av.y, scale[ka + 1], shift[ka + 1]), 0.0f);
    } else {
      a.x = av.x; a.y = av.y;
    }
    b.x = W[(size_t)ka * Cout + colB];
    b.y = W[(size_t)(ka + 1) * Cout + colB];
    // v_wmma_f32_16x16x4_f32: (neg_a, A, neg_b, B, c_mod, C, reuse_a, reuse_b)
    c = __builtin_amdgcn_wmma_f32_16x16x4_f32(false, a, false, b, (short)0, c, false, false);
  }

  const float bb = bias[colB];
#pragma unroll
  for (int r = 0; r < 8; ++r) {
    int row = tileM + r + half * 8;
    float v = c[r] + bb;
    if (POSTRELU) v = fmaxf(v, 0.0f);
    D[(size_t)row * Cout + colB] = v;
  }
}

// ---------------- column sum / sumsq for BatchNorm stats ----------------
// blockDim.x = 256; thread owns channel tid % C; 256/C rows per block-iteration.

__global__ void k_col_stats(const float* __restrict__ H, float* __restrict__ sums,
                            float* __restrict__ sumsq, int nRows, int C) {
  __shared__ float ls[256];
  __shared__ float ls2[256];
  const int tid = threadIdx.x;
  const int c = tid & (C - 1);
  const int rsub = tid / C;
  const int rowsPerIter = blockDim.x / C;
  float s = 0.0f, s2 = 0.0f;
  for (int row = blockIdx.x * rowsPerIter + rsub; row < nRows; row += gridDim.x * rowsPerIter) {
    float v = H[(size_t)row * C + c];
    s += v;
    s2 = fmaf(v, v, s2);
  }
  ls[tid] = s;
  ls2[tid] = s2;
  __syncthreads();
  if (tid < C) {   // rsub == 0, c == tid
    for (int j = 1; j < rowsPerIter; ++j) {
      s  += ls[tid + j * C];
      s2 += ls2[tid + j * C];
    }
    atomicAdd(&sums[c], s);
    atomicAdd(&sumsq[c], s2);
  }
}

__global__ void k_bn_finalize(const float* __restrict__ sums, const float* __restrict__ sumsq,
                              const float* __restrict__ gamma, const float* __restrict__ beta,
                              float* __restrict__ scale, float* __restrict__ shift,
                              float invN, int C) {
  int c = blockIdx.x * blockDim.x + threadIdx.x;
  if (c >= C) return;
  float mean = sums[c] * invN;
  float var  = sumsq[c] * invN - mean * mean;   // biased variance (torch BN)
  float sc = gamma[c] * rsqrtf(var + 1e-5f);
  scale[c] = sc;
  shift[c] = fmaf(-mean, sc, beta[c]);
}

// ---------------- log_softmax over rows of 64 (wave per row, wave32) ----------------

__global__ void k_log_softmax64(float* __restrict__ out, int nRows) {
  int row  = (blockIdx.x * blockDim.x + threadIdx.x) >> 5;
  int lane = threadIdx.x & 31;
  if (row >= nRows) return;
  float* p = out + (size_t)row * 64;
  float v0 = p[lane], v1 = p[lane + 32];
  float m = fmaxf(v0, v1);
#pragma unroll
  for (int off = 16; off > 0; off >>= 1) m = fmaxf(m, __shfl_xor(m, off, 32));
  float e = expf(v0 - m) + expf(v1 - m);
#pragma unroll
  for (int off = 16; off > 0; off >>= 1) e += __shfl_xor(e, off, 32);
  float lse = m + logf(e);
  p[lane]      = v0 - lse;
  p[lane + 32] = v1 - lse;
}

// ---------------- driver ----------------

extern "C" void kernel_launch(void* const* d_in, const int* in_sizes, int n_in,
                              void* d_out, int out_size, void* d_ws, size_t ws_size,
                              hipStream_t stream) {
  const float* x   = (const float*)d_in[0];
  const int*   ei  = (const int*)d_in[1];
  const float* w1a = (const float*)d_in[2];
  const float* b1a = (const float*)d_in[3];
  const float* g1  = (const float*)d_in[4];
  const float* be1 = (const float*)d_in[5];
  const float* w1b = (const float*)d_in[6];
  const float* b1b = (const float*)d_in[7];
  const float* w2a = (const float*)d_in[8];
  const float* b2a = (const float*)d_in[9];
  const float* g2  = (const float*)d_in[10];
  const float* be2 = (const float*)d_in[11];
  const float* w2b = (const float*)d_in[12];
  const float* b2b = (const float*)d_in[13];

  const int* srcIdx = ei;             // edge_index[0]
  const int* dstIdx = ei + kEdges;    // edge_index[1]

  float* agg   = (float*)d_ws;                      // 50000*128
  float* hbuf  = agg  + (size_t)kNodes * 128;       // 50000*128
  float* mid   = hbuf + (size_t)kNodes * 128;       // 50000*128
  float* sums  = mid  + (size_t)kNodes * 128;       // 128
  float* sumsq = sums + 128;                        // 128
  float* scale = sumsq + 128;                       // 128
  float* shift = scale + 128;                       // 128
  float* outp  = (float*)d_out;                     // 50000*64

  const int n128 = kNodes * 128;
  const float invN = 1.0f / (float)kNodes;
  const unsigned scatterBlocks = (unsigned)(((long long)kEdges * 32 + 255) / 256);

  // ---------------- layer 1 ----------------
  k_copy4<<<(n128 / 4 + 255) / 256, 256, 0, stream>>>((const float4*)x, (float4*)agg, n128 / 4);
  k_scatter_add<<<scatterBlocks, 256, 0, stream>>>(x, agg, srcIdx, dstIdx, kEdges, 128);
  k_gemm_wmma<0, 0><<<kNodes / 16, 256, 0, stream>>>(agg, w1a, b1a, nullptr, nullptr, hbuf, 128, 128);
  k_zero<<<1, 256, 0, stream>>>(sums, 256);
  k_col_stats<<<512, 256, 0, stream>>>(hbuf, sums, sumsq, kNodes, 128);
  k_bn_finalize<<<1, 128, 0, stream>>>(sums, sumsq, g1, be1, scale, shift, invN, 128);
  // h = relu( relu(bn(h)) @ w1b + b1b )  : BN+ReLU fused as A pre-op, ReLU epilogue
  k_gemm_wmma<1, 1><<<kNodes / 16, 256, 0, stream>>>(hbuf, w1b, b1b, scale, shift, mid, 128, 128);

  // ---------------- layer 2 ----------------
  k_copy4<<<(n128 / 4 + 255) / 256, 256, 0, stream>>>((const float4*)mid, (float4*)agg, n128 / 4);
  k_scatter_add<<<scatterBlocks, 256, 0, stream>>>(mid, agg, srcIdx, dstIdx, kEdges, 128);
  k_gemm_wmma<0, 0><<<kNodes / 16, 128, 0, stream>>>(agg, w2a, b2a, nullptr, nullptr, hbuf, 128, 64);
  k_zero<<<1, 256, 0, stream>>>(sums, 256);
  k_col_stats<<<512, 256, 0, stream>>>(hbuf, sums, sumsq, kNodes, 64);
  k_bn_finalize<<<1, 64, 0, stream>>>(sums, sumsq, g2, be2, scale, shift, invN, 64);
  k_gemm_wmma<1, 0><<<kNodes / 16, 128, 0, stream>>>(hbuf, w2b, b2b, scale, shift, outp, 64, 64);

  k_log_softmax64<<<(kNodes * 32 + 255) / 256, 256, 0, stream>>>(outp, kNodes);
}